// VectorQuantiser_14637248545124
// MI455X (gfx1250) — compile-verified
//
#include <hip/hip_runtime.h>
#include <math.h>

typedef __attribute__((ext_vector_type(2))) float v2f;
typedef __attribute__((ext_vector_type(8))) float v8f;

#define K_CODES        1024
#define DIM            64
#define BETA           0.25f
#define CHUNK_CODES    256
#define CB_STRIDE      68          // 64 + 4 pad floats: bank-conflict-free B reads
#define ROWS_PER_BLOCK 128
#define WAVES          8

__global__ __launch_bounds__(256)
void vq_wmma_kernel(const float* __restrict__ z,
                    const float* __restrict__ cb,
                    int*   __restrict__ e_out,
                    float* __restrict__ zq_out,
                    float* __restrict__ sim_out,
                    float* __restrict__ loss_out)
{
    __shared__ float lds_cb[CHUNK_CODES * CB_STRIDE]; // 68 KB codebook chunk
    __shared__ float lds_ne[K_CODES];                 // ||e||^2
    __shared__ float lds_re[K_CODES];                 // rsqrt(||e||^2)
    __shared__ float lds_nz[ROWS_PER_BLOCK];          // ||z||^2 per row
    __shared__ int   lds_bi[ROWS_PER_BLOCK];          // argmin per row

    const unsigned tid  = threadIdx.x;
    const unsigned lane = tid & 31u;
    const unsigned wave = tid >> 5;
    const unsigned half = lane >> 4;     // 0: lanes 0-15, 1: lanes 16-31
    const unsigned l15  = lane & 15u;

    // ---- codebook norms (once per block) ----
    for (unsigned c = tid; c < K_CODES; c += 256) {
        const float4* p = (const float4*)(cb + (size_t)c * DIM);
        float s = 0.f;
        #pragma unroll
        for (int i = 0; i < 16; ++i) {
            float4 q = p[i];
            s += q.x*q.x + q.y*q.y + q.z*q.z + q.w*q.w;
        }
        lds_ne[c] = s;
        lds_re[c] = rsqrtf(s);
    }

    // ---- A fragments: 16 rows x 64, ISA 16x4 f32 layout per k-chunk ----
    // lane l (l<16): row m0+l, elems {4kc, 4kc+1}; lane l+16: elems {4kc+2, 4kc+3}
    const unsigned m0 = (blockIdx.x * WAVES + wave) * 16u;
    v2f afrag[16];
    float nzpart = 0.f;
    const float* zrow = z + (size_t)(m0 + l15) * DIM;
    #pragma unroll
    for (int kc = 0; kc < 16; ++kc) {
        unsigned d = 4u*(unsigned)kc + 2u*half;
        float x = zrow[d], y = zrow[d + 1];
        afrag[kc].x = x; afrag[kc].y = y;
        nzpart += x*x + y*y;
    }
    float nzrow = nzpart + __shfl_xor(nzpart, 16, 32);   // full ||z||^2 of row m0+l15
    if (half == 0) lds_nz[wave * 16u + l15] = nzrow;
    __syncthreads();

    // per C-layout rows (VGPR v <-> row v + 8*half)
    float nzr[8], rz[8];
    #pragma unroll
    for (int v = 0; v < 8; ++v) {
        float t = lds_nz[wave * 16u + (unsigned)v + 8u*half];
        nzr[v] = t;
        rz[v]  = rsqrtf(t);
    }

    float bv[8]; int bi[8];
    #pragma unroll
    for (int v = 0; v < 8; ++v) { bv[v] = 3.4e38f; bi[v] = 0; }

    for (int chunk = 0; chunk < K_CODES / CHUNK_CODES; ++chunk) {
        __syncthreads();
        // stage 256 codes (64 KB) into LDS with stride-68 padding
        {
            const float4* src = (const float4*)(cb + (size_t)chunk * CHUNK_CODES * DIM);
            for (unsigned i = tid; i < CHUNK_CODES * 16u; i += 256u) {
                unsigned code = i >> 4, f4 = i & 15u;
                float4 q = src[code * 16u + f4];
                float* dst = &lds_cb[code * CB_STRIDE + f4 * 4u];
                dst[0] = q.x; dst[1] = q.y; dst[2] = q.z; dst[3] = q.w;
            }
        }
        __syncthreads();

        // prefetch next chunk into cache while this chunk computes
        if (chunk + 1 < K_CODES / CHUNK_CODES) {
            const float* nxt = cb + (size_t)(chunk + 1) * CHUNK_CODES * DIM;
            __builtin_prefetch(nxt + (size_t)tid * 64u, 0, 1);
            __builtin_prefetch(nxt + (size_t)tid * 64u + 32u, 0, 1);
        }

        #pragma unroll 1
        for (int t = 0; t < CHUNK_CODES / 16; ++t) {
            const unsigned n0   = (unsigned)chunk * CHUNK_CODES + (unsigned)t * 16u;
            const unsigned cidx = (unsigned)t * 16u + l15;     // code within chunk
            const float* bbase  = &lds_cb[cidx * CB_STRIDE + 2u*half];

            // preload all B fragments for this tile (one dscnt wait, then
            // WMMAs issue back-to-back)
            v2f bfrag[16];
            #pragma unroll
            for (int kc = 0; kc < 16; ++kc) {
                bfrag[kc].x = bbase[4*kc];
                bfrag[kc].y = bbase[4*kc + 1];
            }

            // two interleaved accumulator chains -> independent WMMA streams
            v8f acc0 = {0.f,0.f,0.f,0.f,0.f,0.f,0.f,0.f};
            v8f acc1 = {0.f,0.f,0.f,0.f,0.f,0.f,0.f,0.f};
            #pragma unroll
            for (int kc = 0; kc < 8; ++kc) {
                acc0 = __builtin_amdgcn_wmma_f32_16x16x4_f32(
                        false, afrag[2*kc],     false, bfrag[2*kc],
                        (short)0, acc0, false, false);
                acc1 = __builtin_amdgcn_wmma_f32_16x16x4_f32(
                        false, afrag[2*kc + 1], false, bfrag[2*kc + 1],
                        (short)0, acc1, false, false);
            }

            const unsigned n  = n0 + l15;
            const float    ne = lds_ne[n];
            const float    re = lds_re[n];
            #pragma unroll
            for (int v = 0; v < 8; ++v) {
                float dotv = acc0[v] + acc1[v];
                float key  = fmaf(-2.f, dotv, ne);             // dist - ||z||^2
                if (key < bv[v]) { bv[v] = key; bi[v] = (int)n; }
                unsigned row = m0 + (unsigned)v + 8u*half;
                __builtin_nontemporal_store(dotv * rz[v] * re,
                                            &sim_out[(size_t)row * K_CODES + n]);
            }
        }
    }

    // ---- argmin reduction across the 16 lanes of each half ----
    #pragma unroll
    for (int v = 0; v < 8; ++v) {
        float mybv = bv[v]; int mybi = bi[v];
        #pragma unroll
        for (int off = 1; off < 16; off <<= 1) {
            float ov = __shfl_xor(mybv, off, 32);
            int   oi = __shfl_xor(mybi, off, 32);
            if (ov < mybv || (ov == mybv && oi < mybi)) { mybv = ov; mybi = oi; }
        }
        if (l15 == 0) {
            unsigned row = m0 + (unsigned)v + 8u*half;
            e_out[row] = mybi;
            float dist = nzr[v] + mybv;                        // ||z - e||^2
            loss_out[row] = sqrtf(fmaxf(dist, 0.f)) * (1.f + BETA);
            lds_bi[wave * 16u + (unsigned)v + 8u*half] = mybi;
        }
    }
    __syncthreads();

    // ---- z_q gather: 128 rows x 16 float4 ----
    const unsigned rowbase = blockIdx.x * ROWS_PER_BLOCK;
    for (unsigned i = tid; i < ROWS_PER_BLOCK * 16u; i += 256u) {
        unsigned r = i >> 4, f4 = i & 15u;
        int idx = lds_bi[r];
        float4 q = ((const float4*)cb)[(size_t)idx * 16u + f4];
        ((float4*)zq_out)[(size_t)(rowbase + r) * 16u + f4] = q;
    }
}

extern "C" void kernel_launch(void* const* d_in, const int* in_sizes, int n_in,
                              void* d_out, int out_size, void* d_ws, size_t ws_size,
                              hipStream_t stream) {
    (void)in_sizes; (void)n_in; (void)d_ws; (void)ws_size; (void)out_size;
    const float* z  = (const float*)d_in[0];   // (16, 8192, 64) f32
    const float* cb = (const float*)d_in[1];   // (1024, 64)     f32

    const size_t BT = 16u * 8192u;             // 131072 rows
    float* out      = (float*)d_out;
    int*   e_out    = (int*)out;                       // BT ints
    float* zq_out   = out + BT;                        // BT*64
    float* sim_out  = zq_out + BT * (size_t)DIM;       // BT*1024
    float* loss_out = sim_out + BT * (size_t)K_CODES;  // BT

    dim3 grid((unsigned)(BT / ROWS_PER_BLOCK));        // 1024 blocks
    dim3 block(256);
    vq_wmma_kernel<<<grid, block, 0, stream>>>(z, cb, e_out, zq_out, sim_out, loss_out);
}